// ModelNew_3556232921787
// MI455X (gfx1250) — compile-verified
//
#include <hip/hip_runtime.h>
#include <hip/hip_bf16.h>
#include <math.h>

// Shapes (hardcoded to the reference):
//   x: (32, 8, 512, 512) f32, weight: (16, 8, 3, 3) f32, bias: (16,) f32
//   out: (32,) f32
// Math: pooled[n,co] = (S_x[n,:] . S_w[co,:]) / (514*514) + bias[co]
//       out[n] = 10 * logsumexp_co(pooled[n,:])

typedef float v2f __attribute__((ext_vector_type(2)));
typedef float v8f __attribute__((ext_vector_type(8)));

#define N_IMG      256        // B * C_in = 32 * 8
#define IMG_ELEMS  262144     // 512 * 512 floats per (n, ci) image
#define CHUNKS     64         // partial sums per image
#define VEC4_PER_CHUNK 1024   // (IMG_ELEMS/4) / CHUNKS
#define THREADS    256

// ---------------------------------------------------------------------------
// Stage 1: streaming reduction of x. Each block reduces one contiguous chunk
// of one (n, ci) image. The wave-wide summation is performed by
// V_WMMA_F32_16X16X4_F32 with an all-ones B matrix: D accumulates row sums of
// A replicated across 16 columns, so sum(D over all lanes/components) equals
// 16 * sum(all A elements fed by the wave). The block then tree-reduces the
// per-lane D fragments in LDS and stores one deterministic partial.
// ---------------------------------------------------------------------------
__global__ __launch_bounds__(THREADS)
void ctc_reduce_x_wmma(const float* __restrict__ x, float* __restrict__ partials)
{
    const int g     = blockIdx.y;   // image index (n*8 + ci), 0..255
    const int chunk = blockIdx.x;   // 0..63
    const int tid   = threadIdx.x;  // 0..255

    const float4* img = reinterpret_cast<const float4*>(x + (size_t)g * IMG_ELEMS);
    const size_t base = (size_t)chunk * VEC4_PER_CHUNK;

    v8f acc0 = {0.f, 0.f, 0.f, 0.f, 0.f, 0.f, 0.f, 0.f};
    v8f acc1 = {0.f, 0.f, 0.f, 0.f, 0.f, 0.f, 0.f, 0.f};
    const v2f ones = {1.0f, 1.0f};

    #pragma unroll
    for (int it = 0; it < 4; ++it) {
        float4 v = img[base + (size_t)it * THREADS + tid];
        v2f a0 = {v.x, v.y};
        v2f a1 = {v.z, v.w};
        // D = A x ones + C  (A: 16x4 f32 tile striped across the wave)
        acc0 = __builtin_amdgcn_wmma_f32_16x16x4_f32(
            false, a0, false, ones, (short)0, acc0, false, false);
        acc1 = __builtin_amdgcn_wmma_f32_16x16x4_f32(
            false, a1, false, ones, (short)0, acc1, false, false);
    }

    float s = (acc0[0] + acc0[1]) + (acc0[2] + acc0[3])
            + (acc0[4] + acc0[5]) + (acc0[6] + acc0[7])
            + (acc1[0] + acc1[1]) + (acc1[2] + acc1[3])
            + (acc1[4] + acc1[5]) + (acc1[6] + acc1[7]);

    __shared__ float red[THREADS];
    red[tid] = s;
    __syncthreads();
    #pragma unroll
    for (int off = THREADS / 2; off > 0; off >>= 1) {
        if (tid < off) red[tid] += red[tid + off];
        __syncthreads();
    }
    if (tid == 0) {
        // Each x element appears 16x in the accumulated D fragments.
        partials[g * CHUNKS + chunk] = red[0] * 0.0625f;
    }
}

// ---------------------------------------------------------------------------
// Stage 2: single block. Deterministically folds the 64 partials per image,
// builds the 3x3-summed kernel S_w, does the 32x16x8 micro-matmul, bias add,
// and the 16-wide logsumexp per batch row.
// ---------------------------------------------------------------------------
__global__ __launch_bounds__(THREADS)
void ctc_finalize(const float* __restrict__ partials,
                  const float* __restrict__ weight,
                  const float* __restrict__ bias,
                  float* __restrict__ out)
{
    __shared__ float s_Sx[N_IMG];    // [n*8 + ci]
    __shared__ float s_Sw[16 * 8];   // [co*8 + ci]

    const int tid = threadIdx.x;

    // S_x: fixed-order sum of 64 partials per image.
    {
        const float* p = partials + (size_t)tid * CHUNKS;
        float s = 0.f;
        #pragma unroll
        for (int c = 0; c < CHUNKS; ++c) s += p[c];
        s_Sx[tid] = s;
    }

    // S_w: sum each 3x3 kernel (16*8 = 128 of them).
    if (tid < 128) {
        const float* w = weight + (size_t)tid * 9;
        float s = 0.f;
        #pragma unroll
        for (int k = 0; k < 9; ++k) s += w[k];
        s_Sw[tid] = s;
    }
    __syncthreads();

    if (tid < 32) {
        const int n = tid;
        const float invArea = 1.0f / (514.0f * 514.0f);
        float p[16];
        float m = -INFINITY;
        #pragma unroll
        for (int co = 0; co < 16; ++co) {
            float dot = 0.f;
            #pragma unroll
            for (int ci = 0; ci < 8; ++ci)
                dot = fmaf(s_Sx[n * 8 + ci], s_Sw[co * 8 + ci], dot);
            p[co] = dot * invArea + bias[co];
            m = fmaxf(m, p[co]);
        }
        float se = 0.f;
        #pragma unroll
        for (int co = 0; co < 16; ++co) se += expf(p[co] - m);
        out[n] = 10.0f * (logf(se) + m);
    }
}

extern "C" void kernel_launch(void* const* d_in, const int* in_sizes, int n_in,
                              void* d_out, int out_size, void* d_ws, size_t ws_size,
                              hipStream_t stream)
{
    const float* x      = (const float*)d_in[0];
    const float* weight = (const float*)d_in[1];
    const float* bias   = (const float*)d_in[2];
    float* out          = (float*)d_out;
    float* partials     = (float*)d_ws;   // N_IMG * CHUNKS = 16384 floats

    (void)in_sizes; (void)n_in; (void)out_size; (void)ws_size;

    ctc_reduce_x_wmma<<<dim3(CHUNKS, N_IMG), THREADS, 0, stream>>>(x, partials);
    ctc_finalize<<<1, THREADS, 0, stream>>>(partials, weight, bias, out);
}